// GraphSAGE_44693429682814
// MI455X (gfx1250) — compile-verified
//
#include <hip/hip_runtime.h>

// ---------------------------------------------------------------------------
// GraphSAGE (3-layer) for MI455X / gfx1250.
// agg = mean_{e: dst(e)=n} h[src(e)];  out = relu(agg @ Wl + bl + h @ Wr)
// f32 throughout; matrix ops use V_WMMA_F32_16X16X4_F32.
// ---------------------------------------------------------------------------

#define N_NODES 40000
#define N_EDGES 640000
#define DFEAT   128

typedef __attribute__((ext_vector_type(2))) float v2f;
typedef __attribute__((ext_vector_type(8))) float v8f;

// ------------------------------ fill zero ----------------------------------
__global__ __launch_bounds__(256) void k_fill_zero(float* __restrict__ p, int n) {
    int i = blockIdx.x * 256 + threadIdx.x;
    if (i < n) p[i] = 0.0f;
}

// ------------------------ degree count (as f32) ----------------------------
__global__ __launch_bounds__(256) void k_count_edges(const long long* __restrict__ dst,
                                                     float* __restrict__ cnt) {
    int e = blockIdx.x * 256 + threadIdx.x;
    if (e < N_EDGES) {
        int d = (int)dst[e];
        unsafeAtomicAdd(&cnt[d], 1.0f);
    }
}

__global__ __launch_bounds__(256) void k_finalize_inv(float* __restrict__ cnt) {
    int i = blockIdx.x * 256 + threadIdx.x;
    if (i < N_NODES) cnt[i] = 1.0f / fmaxf(cnt[i], 1.0f);
}

// --------------------- edge scatter: agg[dst] += h[src] --------------------
// One wave (32 lanes) per edge: float4 gather (128 floats/edge), 4 f32
// atomics per lane. Gather + RMW stay resident in the 192MB L2.
__global__ __launch_bounds__(256) void k_scatter(const float* __restrict__ h,
                                                 const long long* __restrict__ src,
                                                 const long long* __restrict__ dst,
                                                 float* __restrict__ agg) {
    int e = blockIdx.x * 8 + (threadIdx.x >> 5);
    if (e >= N_EDGES) return;
    int lane = threadIdx.x & 31;
    int s = (int)src[e];
    int d = (int)dst[e];
    const float4* hs = reinterpret_cast<const float4*>(h + (size_t)s * DFEAT);
    float4 v = hs[lane];
    float* ap = agg + (size_t)d * DFEAT + lane * 4;
    unsafeAtomicAdd(ap + 0, v.x);
    unsafeAtomicAdd(ap + 1, v.y);
    unsafeAtomicAdd(ap + 2, v.z);
    unsafeAtomicAdd(ap + 3, v.w);
}

// --------------------- fused SAGE GEMM with f32 WMMA -----------------------
// out[r, c] = relu( sum_k agg[r,k]*inv[r]*Wl[k,c] + bl[c] + sum_k X[r,k]*Wr[k,c] )
// Block = 256 threads = 8 waves; blockIdx.x selects a 16-row slab; wave w
// owns cols [16w, 16w+16). Each wave runs 2 * 32 V_WMMA_F32_16X16X4_F32.
//
// VGPR layouts per CDNA5 ISA 7.12.2 (wave32):
//   A (16x4 f32):  lanes 0-15 -> rows 0-15, VGPR0=K0, VGPR1=K1;
//                  lanes 16-31 -> same rows, VGPR0=K2, VGPR1=K3.
//   B (4x16 f32):  lanes 0-15 -> cols, VGPR0=K0 row, VGPR1=K1 row;
//                  lanes 16-31 -> cols, VGPR0=K2 row, VGPR1=K3 row.
//   C/D (16x16):   VGPR i -> row i (lanes 0-15) / row i+8 (lanes 16-31).
__global__ __launch_bounds__(256) void k_sage_gemm(const float* __restrict__ X,
                                                   const float* __restrict__ agg,
                                                   const float* __restrict__ invc,
                                                   const float* __restrict__ Wl,
                                                   const float* __restrict__ bl,
                                                   const float* __restrict__ Wr,
                                                   float* __restrict__ out,
                                                   int relu) {
    const int lane = threadIdx.x & 31;
    const int wave = threadIdx.x >> 5;
    const int r0   = blockIdx.x * 16;
    const int c0   = wave * 16;
    const int mlo  = lane & 15;   // row index for A, col index for B/C/D
    const int hi   = lane >> 4;   // K-half selector (0 -> K0/K1, 1 -> K2/K3)

    const int row = r0 + mlo;
    const int col = c0 + mlo;
    const float inv = invc[row];
    const float bias = bl[col];

    v8f c;
#pragma unroll
    for (int i = 0; i < 8; ++i) c[i] = bias;

    const float* xrow = X   + (size_t)row * DFEAT;
    const float* arow = agg + (size_t)row * DFEAT;

#pragma unroll 4
    for (int k = 0; k < DFEAT; k += 4) {
        const int ka = k + hi * 2;
        // lin_l on mean-aggregated neighbors
        v2f a1, b1;
        a1.x = arow[ka] * inv;
        a1.y = arow[ka + 1] * inv;
        b1.x = Wl[(size_t)ka * DFEAT + col];
        b1.y = Wl[(size_t)(ka + 1) * DFEAT + col];
        c = __builtin_amdgcn_wmma_f32_16x16x4_f32(false, a1, false, b1,
                                                  (short)0, c, false, false);
        // lin_r on node's own features
        v2f a2, b2;
        a2.x = xrow[ka];
        a2.y = xrow[ka + 1];
        b2.x = Wr[(size_t)ka * DFEAT + col];
        b2.y = Wr[(size_t)(ka + 1) * DFEAT + col];
        c = __builtin_amdgcn_wmma_f32_16x16x4_f32(false, a2, false, b2,
                                                  (short)0, c, false, false);
    }

#pragma unroll
    for (int i = 0; i < 8; ++i) {
        float v = c[i];
        if (relu) v = fmaxf(v, 0.0f);
        const int m = i + hi * 8;
        out[(size_t)(r0 + m) * DFEAT + col] = v;
    }
}

// ---------------------------------------------------------------------------
extern "C" void kernel_launch(void* const* d_in, const int* in_sizes, int n_in,
                              void* d_out, int out_size, void* d_ws, size_t ws_size,
                              hipStream_t stream) {
    const float*     x   = (const float*)d_in[0];
    const long long* ei  = (const long long*)d_in[1];
    // d_in[2] = edge_weight (unused by the reference)
    const float* Wl0 = (const float*)d_in[3];
    const float* bl0 = (const float*)d_in[4];
    const float* Wr0 = (const float*)d_in[5];
    const float* Wl1 = (const float*)d_in[6];
    const float* bl1 = (const float*)d_in[7];
    const float* Wr1 = (const float*)d_in[8];
    const float* Wl2 = (const float*)d_in[9];
    const float* bl2 = (const float*)d_in[10];
    const float* Wr2 = (const float*)d_in[11];

    const long long* src = ei;
    const long long* dst = ei + N_EDGES;

    // workspace layout
    float* agg  = (float*)d_ws;                       // N*128
    float* h1   = agg + (size_t)N_NODES * DFEAT;      // N*128
    float* h2   = h1  + (size_t)N_NODES * DFEAT;      // N*128
    float* invc = h2  + (size_t)N_NODES * DFEAT;      // N

    const int NF = N_NODES * DFEAT;
    dim3 blk(256);
    dim3 gN((N_NODES + 255) / 256);
    dim3 gE((N_EDGES + 255) / 256);
    dim3 gNF((NF + 255) / 256);
    dim3 gScat(N_EDGES / 8);          // 8 edges (waves) per 256-thread block
    dim3 gGemm(N_NODES / 16);         // 40000 / 16 = 2500 row slabs

    float* out = (float*)d_out;

    // in-degree -> 1/max(cnt,1)  (shared by all three layers)
    k_fill_zero<<<gN, blk, 0, stream>>>(invc, N_NODES);
    k_count_edges<<<gE, blk, 0, stream>>>(dst, invc);
    k_finalize_inv<<<gN, blk, 0, stream>>>(invc);

    // layer 0: x -> h1 (relu)
    k_fill_zero<<<gNF, blk, 0, stream>>>(agg, NF);
    k_scatter<<<gScat, blk, 0, stream>>>(x, src, dst, agg);
    k_sage_gemm<<<gGemm, blk, 0, stream>>>(x, agg, invc, Wl0, bl0, Wr0, h1, 1);

    // layer 1: h1 -> h2 (relu)
    k_fill_zero<<<gNF, blk, 0, stream>>>(agg, NF);
    k_scatter<<<gScat, blk, 0, stream>>>(h1, src, dst, agg);
    k_sage_gemm<<<gGemm, blk, 0, stream>>>(h1, agg, invc, Wl1, bl1, Wr1, h2, 1);

    // layer 2: h2 -> out (no relu)
    k_fill_zero<<<gNF, blk, 0, stream>>>(agg, NF);
    k_scatter<<<gScat, blk, 0, stream>>>(h2, src, dst, agg);
    k_sage_gemm<<<gGemm, blk, 0, stream>>>(h2, agg, invc, Wl2, bl2, Wr2, out, 0);
}